// GraphConvModel_17635135718037
// MI455X (gfx1250) — compile-verified
//
#include <hip/hip_runtime.h>

#define NN 100000
#define NE 3200000
#define NG 512
#define BN_EPS 1e-5f

typedef float v2f __attribute__((ext_vector_type(2)));
typedef float v8f __attribute__((ext_vector_type(8)));

// ---------------------------------------------------------------------------
// util: grid-stride fill
// ---------------------------------------------------------------------------
__global__ void fill_f32(float* __restrict__ p, float v, long n) {
    long i = (long)blockIdx.x * blockDim.x + threadIdx.x;
    long stride = (long)gridDim.x * blockDim.x;
    for (; i < n; i += stride) p[i] = v;
}

// ---------------------------------------------------------------------------
// build padded weight [K x 64]: cols 0..29 = W_rel, cols 32..61 = W_root
// ---------------------------------------------------------------------------
__global__ void build_w1(const float* __restrict__ Wr, const float* __restrict__ Wo,
                         float* __restrict__ Wp) {
    int t = blockIdx.x * blockDim.x + threadIdx.x;   // 128*64
    if (t >= 128 * 64) return;
    int k = t >> 6, c = t & 63;
    float v = 0.0f;
    if (c < 30)                 v = Wr[k * 30 + c];
    else if (c >= 32 && c < 62) v = Wo[k * 30 + (c - 32)];
    Wp[t] = v;
}

__global__ void build_w2(const float* __restrict__ Wr, const float* __restrict__ Wo,
                         float* __restrict__ Wp) {
    int t = blockIdx.x * blockDim.x + threadIdx.x;   // 32*64 (rows 30,31 pad = 0)
    if (t >= 32 * 64) return;
    int k = t >> 6, c = t & 63;
    float v = 0.0f;
    if (k < 30) {
        if (c < 20)                 v = Wr[k * 20 + c];
        else if (c >= 32 && c < 52) v = Wo[k * 20 + (c - 32)];
    }
    Wp[t] = v;
}

// ---------------------------------------------------------------------------
// WMMA fp32 GEMM:  Y[N x 64] = A[N x K] @ Wp[K x 64]
// One wave computes a full 16x64 output strip: A fragment loaded once per
// K-step (global b64) and fed to 4 WMMAs against 4 B fragments, each a single
// ds_load_b64 from K-pair-interleaved LDS-staged weights:
//     lds[(k>>1)*128 + c*2 + (k&1)] = Wp[k][c]
// so fragment {Wp[kb][c], Wp[kb+1][c]} is contiguous + 8B aligned.
// 10 waves/block * 625 blocks == 6250 row tiles exactly -> no predication,
// EXEC all-1s for every WMMA.
// ---------------------------------------------------------------------------
__global__ void __launch_bounds__(320) gemm_wmma_f32(const float* __restrict__ A,
                                                     const float* __restrict__ Wp,
                                                     float* __restrict__ Y,
                                                     int K) {
    __shared__ float lds[128 * 64];                  // weights (<= 32 KB)
    for (int i = threadIdx.x; i < K * 64; i += 320) {
        int k = i >> 6, c = i & 63;
        lds[(k >> 1) * 128 + c * 2 + (k & 1)] = Wp[i];
    }
    __syncthreads();

    int mt   = blockIdx.x * 10 + (threadIdx.x >> 5); // row tile
    int lane = threadIdx.x & 31;
    int m    = lane & 15;                            // A row / B,D column
    int kh   = lane >> 4;                            // K-half per ISA layout

    const float* arow = A + (long)(mt * 16 + m) * K;

    v8f c0 = {}, c1 = {}, c2 = {}, c3 = {};
#pragma unroll 4
    for (int k0 = 0; k0 < K; k0 += 4) {
        int kb = k0 + kh * 2;                        // even
        v2f a = *(const v2f*)(arow + kb);            // A VGPR0/1: K = kb, kb+1
        const float* w = &lds[(kb >> 1) * 128 + m * 2];
        v2f b0 = *(const v2f*)(w + 0);               // cols  0..15
        v2f b1 = *(const v2f*)(w + 32);              // cols 16..31
        v2f b2 = *(const v2f*)(w + 64);              // cols 32..47
        v2f b3 = *(const v2f*)(w + 96);              // cols 48..63
        c0 = __builtin_amdgcn_wmma_f32_16x16x4_f32(false, a, false, b0, (short)0, c0, false, false);
        c1 = __builtin_amdgcn_wmma_f32_16x16x4_f32(false, a, false, b1, (short)0, c1, false, false);
        c2 = __builtin_amdgcn_wmma_f32_16x16x4_f32(false, a, false, b2, (short)0, c2, false, false);
        c3 = __builtin_amdgcn_wmma_f32_16x16x4_f32(false, a, false, b3, (short)0, c3, false, false);
    }

    // D layout: VGPR i -> row mt*16 + i + kh*8, col (tile base) + m
    float* yrow = Y + (long)(mt * 16 + kh * 8) * 64 + m;
#pragma unroll
    for (int i = 0; i < 8; ++i) {
        yrow[(long)i * 64 + 0]  = c0[i];
        yrow[(long)i * 64 + 16] = c1[i];
        yrow[(long)i * 64 + 32] = c2[i];
        yrow[(long)i * 64 + 48] = c3[i];
    }
}

// ---------------------------------------------------------------------------
// edge scatter: AGG[dst, c] += Y[src, c]  (one wave per edge, lane = column)
// Y row stride 64 (projected features in cols 0..ncols-1), AGG row stride 32
// ---------------------------------------------------------------------------
__global__ void __launch_bounds__(256) edge_scatter(const int* __restrict__ src,
                                                    const int* __restrict__ dst,
                                                    const float* __restrict__ Y,
                                                    float* __restrict__ AGG,
                                                    int nE, int ncols) {
    int e = blockIdx.x * 8 + (threadIdx.x >> 5);
    if (e >= nE) return;
    int c = threadIdx.x & 31;
    int s = src[e];
    int d = dst[e];
    if (c < ncols) {
        float v = Y[(long)s * 64 + c];
        atomicAdd(&AGG[(long)d * 32 + c], v);
    }
}

// ---------------------------------------------------------------------------
// node update 1: H1 = BN(relu(AGG1 + y_root + b)), padded to 32 cols
// ---------------------------------------------------------------------------
__global__ void node_update1(const float* __restrict__ AGG1, const float* __restrict__ Y1,
                             const float* __restrict__ b1,
                             const float* __restrict__ g, const float* __restrict__ bt,
                             const float* __restrict__ mean, const float* __restrict__ var,
                             float* __restrict__ H1) {
    int t = blockIdx.x * blockDim.x + threadIdx.x;   // NN*32
    if (t >= NN * 32) return;
    int node = t >> 5, c = t & 31;
    float v = 0.0f;
    if (c < 30) {
        v = AGG1[t] + Y1[(long)node * 64 + 32 + c] + b1[c];
        v = fmaxf(v, 0.0f);
        v = (v - mean[c]) * rsqrtf(var[c] + BN_EPS) * g[c] + bt[c];
    }
    H1[t] = v;
}

// ---------------------------------------------------------------------------
// float atomic max via signed/unsigned int trick (init = -inf)
// ---------------------------------------------------------------------------
__device__ __forceinline__ void atomicMaxF32(float* addr, float val) {
    if (val >= 0.0f)
        atomicMax((int*)addr, __float_as_int(val));
    else
        atomicMin((unsigned int*)addr, __float_as_uint(val));
}

// ---------------------------------------------------------------------------
// node update 2 + fused global pooling (max / sum / count)
// ---------------------------------------------------------------------------
__global__ void node_update2_pool(const float* __restrict__ AGG2, const float* __restrict__ Y2,
                                  const float* __restrict__ b2,
                                  const float* __restrict__ g, const float* __restrict__ bt,
                                  const float* __restrict__ mean, const float* __restrict__ var,
                                  const int* __restrict__ batch,
                                  float* __restrict__ gmax, float* __restrict__ gsum,
                                  float* __restrict__ cnt) {
    int t = blockIdx.x * blockDim.x + threadIdx.x;   // NN*32
    if (t >= NN * 32) return;
    int node = t >> 5, c = t & 31;
    int gid = batch[node];
    if (c < 20) {
        float v = AGG2[t] + Y2[(long)node * 64 + 32 + c] + b2[c];
        v = fmaxf(v, 0.0f);
        v = (v - mean[c]) * rsqrtf(var[c] + BN_EPS) * g[c] + bt[c];
        atomicMaxF32(&gmax[gid * 32 + c], v);
        atomicAdd(&gsum[gid * 32 + c], v);
    }
    if (c == 0) atomicAdd(&cnt[gid], 1.0f);
}

// ---------------------------------------------------------------------------
// head: z=[gmax|gmean] (G x 40) -> relu(z@W1+b1) -> sigmoid(h@W2+b2)
// ---------------------------------------------------------------------------
__global__ void head_kernel(const float* __restrict__ gmax, const float* __restrict__ gsum,
                            const float* __restrict__ cnt,
                            const float* __restrict__ W1, const float* __restrict__ b1,
                            const float* __restrict__ W2, const float* __restrict__ b2,
                            float* __restrict__ out) {
    int gI = blockIdx.x * blockDim.x + threadIdx.x;
    if (gI >= NG) return;
    float z[40];
    float invc = 1.0f / fmaxf(cnt[gI], 1.0f);
    for (int c = 0; c < 20; ++c) {
        float m = gmax[gI * 32 + c];
        z[c]      = (m < -1.0e30f) ? 0.0f : m;   // isfinite guard (init was -inf)
        z[20 + c] = gsum[gI * 32 + c] * invc;
    }
    float h[10];
#pragma unroll
    for (int j = 0; j < 10; ++j) {
        float acc = b1[j];
        for (int i = 0; i < 40; ++i) acc += z[i] * W1[i * 10 + j];
        h[j] = fmaxf(acc, 0.0f);
    }
    float o = b2[0];
#pragma unroll
    for (int j = 0; j < 10; ++j) o += h[j] * W2[j];
    out[gI] = 1.0f / (1.0f + __expf(-o));
}

// ---------------------------------------------------------------------------
extern "C" void kernel_launch(void* const* d_in, const int* in_sizes, int n_in,
                              void* d_out, int out_size, void* d_ws, size_t ws_size,
                              hipStream_t stream) {
    const float* x       = (const float*)d_in[0];
    const int*   ei      = (const int*)d_in[1];          // [2, NE]
    const int*   batch   = (const int*)d_in[2];
    const float* W_rel1  = (const float*)d_in[3];
    const float* W_root1 = (const float*)d_in[4];
    const float* b1      = (const float*)d_in[5];
    const float* bn1_g   = (const float*)d_in[6];
    const float* bn1_b   = (const float*)d_in[7];
    const float* bn1_m   = (const float*)d_in[8];
    const float* bn1_v   = (const float*)d_in[9];
    const float* W_rel2  = (const float*)d_in[10];
    const float* W_root2 = (const float*)d_in[11];
    const float* b2      = (const float*)d_in[12];
    const float* bn2_g   = (const float*)d_in[13];
    const float* bn2_b   = (const float*)d_in[14];
    const float* bn2_m   = (const float*)d_in[15];
    const float* bn2_v   = (const float*)d_in[16];
    const float* W_lin1  = (const float*)d_in[17];
    const float* b_lin1  = (const float*)d_in[18];
    const float* W_lin2  = (const float*)d_in[19];
    const float* b_lin2  = (const float*)d_in[20];

    const int* src = ei;
    const int* dst = ei + NE;

    // workspace layout (floats)
    float* ws   = (float*)d_ws;
    float* Wp1  = ws;                       // 128*64      = 8192
    float* Wp2  = Wp1 + 128 * 64;           // 32*64       = 2048
    float* Y1   = Wp2 + 32 * 64;            // NN*64
    float* Y2   = Y1 + (long)NN * 64;       // NN*64
    float* H1   = Y2 + (long)NN * 64;       // NN*32
    float* AGG1 = H1 + (long)NN * 32;       // NN*32  --+
    float* AGG2 = AGG1 + (long)NN * 32;     // NN*32    | contiguous zero region
    float* gsum = AGG2 + (long)NN * 32;     // NG*32    |
    float* cnt  = gsum + NG * 32;           // NG     --+
    float* gmax = cnt + NG;                 // NG*32  (init -inf)

    // 1) init accumulators
    long zeroN = (long)NN * 32 * 2 + NG * 32 + NG;   // AGG1,AGG2,gsum,cnt
    fill_f32<<<2048, 256, 0, stream>>>(AGG1, 0.0f, zeroN);
    fill_f32<<<64, 256, 0, stream>>>(gmax, -__builtin_inff(), (long)NG * 32);
    build_w1<<<(128 * 64 + 255) / 256, 256, 0, stream>>>(W_rel1, W_root1, Wp1);
    build_w2<<<(32 * 64 + 255) / 256, 256, 0, stream>>>(W_rel2, W_root2, Wp2);

    // 2) layer-1 projection (WMMA): Y1 = x @ [W_rel1 | W_root1]   (K=128)
    gemm_wmma_f32<<<NN / 16 / 10, 320, 0, stream>>>(x, Wp1, Y1, 128);

    // 3) edge scatter in projected (30-dim) space
    edge_scatter<<<NE / 8, 256, 0, stream>>>(src, dst, Y1, AGG1, NE, 30);

    // 4) bias + relu + BN -> H1 (padded N x 32)
    node_update1<<<(NN * 32 + 255) / 256, 256, 0, stream>>>(AGG1, Y1, b1, bn1_g, bn1_b,
                                                            bn1_m, bn1_v, H1);

    // 5) layer-2 projection (WMMA): Y2 = H1 @ [W_rel2 | W_root2]  (K=32)
    gemm_wmma_f32<<<NN / 16 / 10, 320, 0, stream>>>(H1, Wp2, Y2, 32);

    // 6) edge scatter in 20-dim space
    edge_scatter<<<NE / 8, 256, 0, stream>>>(src, dst, Y2, AGG2, NE, 20);

    // 7) bias + relu + BN + fused pooling
    node_update2_pool<<<(NN * 32 + 255) / 256, 256, 0, stream>>>(AGG2, Y2, b2, bn2_g, bn2_b,
                                                                 bn2_m, bn2_v, batch,
                                                                 gmax, gsum, cnt);

    // 8) MLP head + sigmoid
    head_kernel<<<(NG + 255) / 256, 256, 0, stream>>>(gmax, gsum, cnt, W_lin1, b_lin1,
                                                      W_lin2, b_lin2, (float*)d_out);
}